// RPNPostProcessor_42606075576323
// MI455X (gfx1250) — compile-verified
//
#include <hip/hip_runtime.h>
#include <hip/hip_bf16.h>
#include <math.h>

// ---------------- problem constants (from reference) ----------------
#define N_IMG   2
#define A_NUM   15
#define H_DIM   128
#define W_DIM   160
#define NA      (A_NUM * H_DIM * W_DIM)   // 307200 anchors per image
#define PRE_N   1000
#define PRE_PAD 1008                       // 63 * 16 (tile-padded)
#define POST_N  300
#define TILE16  63
#define NMS_THR 0.7f

typedef __attribute__((ext_vector_type(2))) float v2f;
typedef __attribute__((ext_vector_type(8))) float v8f;
typedef __attribute__((ext_vector_type(4))) int   v4i;

#if __has_builtin(__builtin_amdgcn_global_load_async_to_lds_b128) && \
    __has_builtin(__builtin_amdgcn_global_load_async_to_lds_b32)
#define HAVE_ASYNC_LDS 1
#else
#define HAVE_ASYNC_LDS 0
#endif

#if HAVE_ASYNC_LDS
// Cast helpers go through integers to avoid generic<->AS pointer casts.
// LDS generic addresses carry the 32-bit LDS offset in the low bits
// (aperture scheme, ISA 10.2); global generic pointers are bit-identical
// to AS1 addresses.
__device__ __forceinline__ __attribute__((address_space(1))) v4i*
glob_v4(const void* p) {
  return (__attribute__((address_space(1))) v4i*)(unsigned long long)p;
}
__device__ __forceinline__ __attribute__((address_space(1))) int*
glob_i32(const void* p) {
  return (__attribute__((address_space(1))) int*)(unsigned long long)p;
}
__device__ __forceinline__ __attribute__((address_space(3))) v4i*
lds_v4(void* p) {
  return (__attribute__((address_space(3))) v4i*)(unsigned)(unsigned long long)p;
}
__device__ __forceinline__ __attribute__((address_space(3))) int*
lds_i32(void* p) {
  return (__attribute__((address_space(3))) int*)(unsigned)(unsigned long long)p;
}
#endif

// ---------------- kernel 1: sigmoid + permute_and_flatten ----------------
// objectness [N,A,H,W] -> scores_flat [N, H*W*A] with f = (h*W+w)*A + a
__global__ __launch_bounds__(256)
void k1_sigmoid(const float* __restrict__ obj, float* __restrict__ scores) {
  int t = blockIdx.x * blockDim.x + threadIdx.x;
  if (t >= N_IMG * NA) return;
  int n  = t / NA;
  int r  = t % NA;
  int a  = r / (H_DIM * W_DIM);
  int hw = r % (H_DIM * W_DIM);
  float x = obj[t];
  float s = 1.0f / (1.0f + expf(-x));
  scores[(size_t)n * NA + (size_t)hw * A_NUM + a] = s;
}

// ---------------- kernel 2: per-image top-1000 (radix select + bitonic) ----
__global__ __launch_bounds__(1024)
void k2_topk(const float* __restrict__ scores, int* __restrict__ selIdx,
             float* __restrict__ selScore) {
  const int n = blockIdx.x;
  const int tid = threadIdx.x;
  const float* sc = scores + (size_t)n * NA;

  __shared__ int hist[256];
  __shared__ unsigned sPrefix;
  __shared__ int sK, cGt, cEq;
  __shared__ unsigned long long comp[1024];

  if (tid == 0) { sPrefix = 0u; sK = PRE_N; }
  __syncthreads();

  // 4 passes of 8-bit radix select (descending) on sortable float keys
  for (int pass = 0; pass < 4; ++pass) {
    const int shift = 24 - 8 * pass;
    for (int i = tid; i < 256; i += 1024) hist[i] = 0;
    __syncthreads();
    const unsigned prefix = sPrefix;
    const unsigned maskHigh = (pass == 0) ? 0u : (0xFFFFFFFFu << (shift + 8));
    for (int i = tid; i < NA; i += 1024) {
      unsigned u = __float_as_uint(sc[i]);
      u = (u & 0x80000000u) ? ~u : (u | 0x80000000u);
      if ((u & maskHigh) == (prefix & maskHigh))
        atomicAdd(&hist[(u >> shift) & 0xFF], 1);
    }
    __syncthreads();
    if (tid == 0) {
      int K = sK;
      int d = 255;
      for (; d > 0; --d) {
        if (K <= hist[d]) break;
        K -= hist[d];
      }
      sPrefix = sPrefix | ((unsigned)d << shift);
      sK = K;
    }
    __syncthreads();
  }

  const unsigned T = sPrefix;      // exact key of the 1000th largest
  const int needEq = sK;           // how many ==T elements we still need
  if (tid == 0) { cGt = 0; cEq = 0; }
  __syncthreads();

  // compact strictly-greater keys
  for (int i = tid; i < NA; i += 1024) {
    unsigned u = __float_as_uint(sc[i]);
    u = (u & 0x80000000u) ? ~u : (u | 0x80000000u);
    if (u > T) {
      int p = atomicAdd(&cGt, 1);
      if (p < 1024)
        comp[p] = ((unsigned long long)u << 32) | (unsigned)(0x7FFFFFFF - i);
    }
  }
  __syncthreads();
  int base = cGt; if (base > 1024) base = 1024;
  // fill remaining slots with ==T keys
  for (int i = tid; i < NA; i += 1024) {
    unsigned u = __float_as_uint(sc[i]);
    u = (u & 0x80000000u) ? ~u : (u | 0x80000000u);
    if (u == T) {
      int p = atomicAdd(&cEq, 1);
      if (p < needEq && (base + p) < 1024)
        comp[base + p] = ((unsigned long long)u << 32) | (unsigned)(0x7FFFFFFF - i);
    }
  }
  __syncthreads();
  int total = base + ((cEq < needEq) ? cEq : needEq);
  if (total > 1024) total = 1024;
  if (tid >= total) comp[tid] = 0ull;
  __syncthreads();

  // bitonic sort 1024 composites, descending (=> score desc, idx asc on ties)
  for (int k = 2; k <= 1024; k <<= 1) {
    for (int j = k >> 1; j > 0; j >>= 1) {
      int i = tid;
      int ixj = i ^ j;
      if (ixj > i) {
        unsigned long long va = comp[i], vb = comp[ixj];
        bool descBlock = ((i & k) == 0);
        bool sw = descBlock ? (va < vb) : (va > vb);
        if (sw) { comp[i] = vb; comp[ixj] = va; }
      }
      __syncthreads();
    }
  }

  if (tid < PRE_N) {
    unsigned long long c = comp[tid];
    int idx = 0x7FFFFFFF - (int)(unsigned)(c & 0xFFFFFFFFu);
    if (idx < 0 || idx >= NA) idx = 0;
    selIdx[n * PRE_N + tid] = idx;
    selScore[n * PRE_N + tid] = sc[idx];
  }
}

// ---------------- kernel 3: gather + rotated box decode + corners ----------
__global__ __launch_bounds__(256)
void k3_decode(const int* __restrict__ selIdx, const float* __restrict__ anchors,
               const float* __restrict__ boxreg, float* __restrict__ props,
               float* __restrict__ corners, float* __restrict__ areas) {
  int t = blockIdx.x * blockDim.x + threadIdx.x;
  if (t >= N_IMG * PRE_PAD) return;
  const int n = t / PRE_PAD, r = t % PRE_PAD;
  float* P = props + (size_t)t * 5;
  float* C = corners + (size_t)t * 8;
  if (r >= PRE_N) {  // tile padding -> degenerate zero boxes
    for (int c = 0; c < 5; ++c) P[c] = 0.f;
    for (int c = 0; c < 8; ++c) C[c] = 0.f;
    areas[t] = 0.f;
    return;
  }
  const int f  = selIdx[n * PRE_N + r];
  const int a  = f % A_NUM;
  const int hw = f / A_NUM;
  const int wc = hw % W_DIM, hr = hw / W_DIM;
  const size_t planeHW = (size_t)H_DIM * W_DIM;
  float rel[5], anc[5];
#pragma unroll
  for (int c = 0; c < 5; ++c)
    rel[c] = boxreg[(((size_t)n * A_NUM + a) * 5 + c) * planeHW + (size_t)hr * W_DIM + wc];
#pragma unroll
  for (int c = 0; c < 5; ++c) anc[c] = anchors[((size_t)n * NA + f) * 5 + c];

  const float CLIP = 4.1351666f;  // log(1000/16)
  const float PI = 3.14159265358979f;
  float pcx = rel[0] * anc[2] + anc[0];
  float pcy = rel[1] * anc[3] + anc[1];
  float pw  = expf(fminf(rel[2], CLIP)) * anc[2];
  float ph  = expf(fminf(rel[3], CLIP)) * anc[3];
  float pa  = rel[4] * (180.0f / PI) + anc[4];
  P[0] = pcx; P[1] = pcy; P[2] = pw; P[3] = ph; P[4] = pa;

  float th = pa * (PI / 180.0f);
  float ct = cosf(th), st = sinf(th);
  float hwd = 0.5f * pw, hht = 0.5f * ph;
  float lx[4] = {-hwd, hwd, hwd, -hwd};
  float ly[4] = {-hht, -hht, hht, hht};
#pragma unroll
  for (int e = 0; e < 4; ++e) {
    C[2 * e]     = pcx + lx[e] * ct - ly[e] * st;
    C[2 * e + 1] = pcy + lx[e] * st + ly[e] * ct;
  }
  areas[t] = pw * ph;
}

// ---------------- kernel 4: tiled pairwise rotated IoU (WMMA union) -------
// One wave32 owns a 16x16 tile. Each lane computes the 8 intersection areas
// of its native WMMA C/D slots (lane<16: M=v,N=lane; lane>=16: M=v+8,N=lane-16),
// then V_WMMA_F32_16X16X4_F32 computes union = area_i + area_j - inter:
//   A(16x4)=[area_row,1,0,0], B(4x16)=[1;area_col;0;0], C=-inter.
// Tile staging goes through GLOBAL_LOAD_ASYNC_TO_LDS (ASYNCcnt).
__global__ __launch_bounds__(256)
void k4_iou(const float* __restrict__ corners, const float* __restrict__ areas,
            float* __restrict__ iouOut) {
  __shared__ __align__(16) float sB[8][288];  // per-wave: rowC[128] colC[128] rowA[16] colA[16]
  const int lane = threadIdx.x & 31;
  const int wv   = threadIdx.x >> 5;
  int tileId = blockIdx.x * 8 + wv;
  const int tilesPerImg = TILE16 * TILE16;
  const int totalTiles  = N_IMG * tilesPerImg;
  const bool validTile = (tileId < totalTiles);
  if (!validTile) tileId = 0;
  const int n  = tileId / tilesPerImg;
  const int tt = tileId % tilesPerImg;
  const int tr = tt / TILE16, tc = tt % TILE16;

  float* rowC = &sB[wv][0];
  float* colC = &sB[wv][128];
  float* rowA = &sB[wv][256];
  float* colA = &sB[wv][272];
  const float* cor = corners + (size_t)n * PRE_PAD * 8;
  const float* are = areas + (size_t)n * PRE_PAD;

  __builtin_prefetch(cor + (size_t)(tr * 16) * 8, 0, 0);  // global_prefetch_b8
  __builtin_prefetch(cor + (size_t)(tc * 16) * 8, 0, 0);

#if HAVE_ASYNC_LDS
  {
    // 512B row-corner block: 32 lanes x 16B; same for col-corner block.
    const float* gRow = cor + (size_t)(tr * 16) * 8 + lane * 4;
    const float* gCol = cor + (size_t)(tc * 16) * 8 + lane * 4;
    __builtin_amdgcn_global_load_async_to_lds_b128(
        glob_v4(gRow), lds_v4(rowC + lane * 4), 0, 0);
    __builtin_amdgcn_global_load_async_to_lds_b128(
        glob_v4(gCol), lds_v4(colC + lane * 4), 0, 0);
    // 32 area floats: lanes 0-15 -> rowA, lanes 16-31 -> colA.
    const float* gA = (lane < 16) ? (are + tr * 16 + lane)
                                  : (are + tc * 16 + (lane - 16));
    float* lA = (lane < 16) ? (rowA + lane) : (colA + (lane - 16));
    __builtin_amdgcn_global_load_async_to_lds_b32(
        glob_i32(gA), lds_i32(lA), 0, 0);
  }
#if __has_builtin(__builtin_amdgcn_s_wait_asynccnt)
  __builtin_amdgcn_s_wait_asynccnt(0);
#else
  asm volatile("s_wait_asynccnt 0" ::: "memory");
#endif
  __syncthreads();
#else
  if (lane < 16) {
    int b = tr * 16 + lane;
#pragma unroll
    for (int q = 0; q < 8; ++q) rowC[lane * 8 + q] = cor[(size_t)b * 8 + q];
    rowA[lane] = are[b];
  } else {
    int l = lane - 16;
    int b = tc * 16 + l;
#pragma unroll
    for (int q = 0; q < 8; ++q) colC[l * 8 + q] = cor[(size_t)b * 8 + q];
    colA[l] = are[b];
  }
  __syncthreads();
#endif

  const int N_ = lane & 15;
  const int mofs = (lane >> 4) * 8;

  // clip rectangle (column box) corners, CCW
  float qx[4], qy[4];
#pragma unroll
  for (int e = 0; e < 4; ++e) {
    qx[e] = colC[N_ * 8 + 2 * e];
    qy[e] = colC[N_ * 8 + 2 * e + 1];
  }

  float inters[8];
  for (int v = 0; v < 8; ++v) {
    const int M = v + mofs;
    float px[16], py[16];
    int cnt = 4;
#pragma unroll
    for (int e = 0; e < 4; ++e) {
      px[e] = rowC[M * 8 + 2 * e];
      py[e] = rowC[M * 8 + 2 * e + 1];
    }
    // Sutherland-Hodgman: clip row rect by the 4 half-planes of col rect
    for (int e = 0; e < 4; ++e) {
      float p0x = qx[e], p0y = qy[e];
      float p1x = qx[(e + 1) & 3], p1y = qy[(e + 1) & 3];
      float ex = p1x - p0x, ey = p1y - p0y;
      float nx2[16], ny2[16];
      int m2 = 0;
      for (int i = 0; i < cnt; ++i) {
        int j = (i + 1 == cnt) ? 0 : i + 1;
        float di = ex * (py[i] - p0y) - ey * (px[i] - p0x);
        float dj = ex * (py[j] - p0y) - ey * (px[j] - p0x);
        bool insI = (di >= 0.0f), insJ = (dj >= 0.0f);
        if (insI) { nx2[m2] = px[i]; ny2[m2] = py[i]; ++m2; }
        if (insI != insJ) {
          float den = di - dj;
          if (fabsf(den) < 1e-8f) den = 1e-8f;
          float ti = di / den;
          nx2[m2] = px[i] + ti * (px[j] - px[i]);
          ny2[m2] = py[i] + ti * (py[j] - py[i]);
          ++m2;
        }
      }
      cnt = (m2 > 8) ? 8 : m2;
      for (int i = 0; i < 8; ++i) {
        px[i] = (i < cnt) ? nx2[i] : 0.f;
        py[i] = (i < cnt) ? ny2[i] : 0.f;
      }
    }
    float s = 0.f;
    for (int i = 0; i < cnt; ++i) {
      int j = (i + 1 == cnt) ? 0 : i + 1;
      s += px[i] * py[j] - px[j] * py[i];
    }
    inters[v] = 0.5f * fabsf(s);
  }

  // union via WMMA: D = A x B + (-inter)  => area_i + area_j - inter
  v2f av, bv;
  av.x = (lane < 16) ? rowA[lane & 15] : 0.0f;  // A K=0 column / K=2 (zero)
  av.y = (lane < 16) ? 1.0f : 0.0f;             // A K=1 column / K=3 (zero)
  bv.x = (lane < 16) ? 1.0f : colA[lane & 15];  // B row K=0 (ones) / K=1 (area_col)
  bv.y = 0.0f;                                  // B rows K=2,K=3
  v8f cacc;
#pragma unroll
  for (int v = 0; v < 8; ++v) cacc[v] = -inters[v];
  v8f dv = __builtin_amdgcn_wmma_f32_16x16x4_f32(
      false, av, false, bv, (short)0, cacc, false, false);
  float dd[8];
#pragma unroll
  for (int v = 0; v < 8; ++v) dd[v] = dv[v];

  if (validTile) {
    float* outp = iouOut + (size_t)n * PRE_PAD * PRE_PAD;
#pragma unroll
    for (int v = 0; v < 8; ++v) {
      int row = tr * 16 + v + mofs;
      int col = tc * 16 + N_;
      float un = dd[v];
      if (un < 1e-8f) un = 1e-8f;
      outp[(size_t)row * PRE_PAD + col] = inters[v] / un;
    }
  }
}

// ---------------- kernel 5: greedy NMS + cap-300 compaction ---------------
__global__ __launch_bounds__(1024)
void k5_nms(const float* __restrict__ iou, const float* __restrict__ props,
            const float* __restrict__ selScore, float* __restrict__ out) {
  const int n = blockIdx.x;
  const int tid = threadIdx.x;
  __shared__ int supp[PRE_N];
  __shared__ int keep[PRE_N];
  __shared__ int pos[PRE_N];
  __shared__ int sTot;

  for (int i = tid; i < PRE_N; i += 1024) supp[i] = 0;
  __syncthreads();

  const float* iouN = iou + (size_t)n * PRE_PAD * PRE_PAD;
  for (int i = 0; i < PRE_N; ++i) {
    int ki = !supp[i];
    if (tid == 0) keep[i] = ki;
    int j = tid;
    if (ki && j < PRE_N && j > i) {
      if (iouN[(size_t)i * PRE_PAD + j] > NMS_THR) supp[j] = 1;
    }
    __syncthreads();
  }

  if (tid == 0) {
    int tot = 0;
    for (int r = 0; r < PRE_N; ++r) {
      if (keep[r] && tot < POST_N) { pos[r] = tot; ++tot; }
      else { keep[r] = 0; pos[r] = -1; }
    }
    sTot = tot;
  }
  __syncthreads();

  float* propsOut = out;                          // [N,300,5]
  float* scrsOut  = out + N_IMG * POST_N * 5;     // [N,300]
  float* validOut = scrsOut + N_IMG * POST_N;     // [N,300]

  for (int s = tid; s < POST_N * 5; s += 1024) propsOut[n * POST_N * 5 + s] = 0.f;
  for (int s = tid; s < POST_N; s += 1024) scrsOut[n * POST_N + s] = 0.f;
  __syncthreads();

  const int tot = sTot;
  for (int r = tid; r < PRE_N; r += 1024) {
    if (keep[r]) {
      int s = pos[r];
#pragma unroll
      for (int c = 0; c < 5; ++c)
        propsOut[(n * POST_N + s) * 5 + c] = props[((size_t)n * PRE_PAD + r) * 5 + c];
      scrsOut[n * POST_N + s] = selScore[n * PRE_N + r];
    }
  }
  for (int s = tid; s < POST_N; s += 1024)
    validOut[n * POST_N + s] = (s < tot) ? 1.0f : 0.0f;
}

// ---------------- launch --------------------------------------------------
extern "C" void kernel_launch(void* const* d_in, const int* in_sizes, int n_in,
                              void* d_out, int out_size, void* d_ws, size_t ws_size,
                              hipStream_t stream) {
  const float* anchors    = (const float*)d_in[0];  // [N, NA, 5]
  const float* objectness = (const float*)d_in[1];  // [N, A, H, W]
  const float* boxreg     = (const float*)d_in[2];  // [N, A*5, H, W]
  float* out = (float*)d_out;

  float* ws = (float*)d_ws;
  size_t off = 0;
  float* scores   = ws + off; off += (size_t)N_IMG * NA;
  int*   selIdx   = (int*)(ws + off); off += (size_t)N_IMG * PRE_N;
  float* selScore = ws + off; off += (size_t)N_IMG * PRE_N;
  float* props    = ws + off; off += (size_t)N_IMG * PRE_PAD * 5;
  float* corners  = ws + off; off += (size_t)N_IMG * PRE_PAD * 8;
  float* areas    = ws + off; off += (size_t)N_IMG * PRE_PAD;
  float* iou      = ws + off; off += (size_t)N_IMG * PRE_PAD * PRE_PAD;

  const int tot1 = N_IMG * NA;
  k1_sigmoid<<<(tot1 + 255) / 256, 256, 0, stream>>>(objectness, scores);
  k2_topk<<<N_IMG, 1024, 0, stream>>>(scores, selIdx, selScore);
  const int tot3 = N_IMG * PRE_PAD;
  k3_decode<<<(tot3 + 255) / 256, 256, 0, stream>>>(selIdx, anchors, boxreg,
                                                    props, corners, areas);
  const int totalTiles = N_IMG * TILE16 * TILE16;
  k4_iou<<<(totalTiles + 7) / 8, 256, 0, stream>>>(corners, areas, iou);
  k5_nms<<<N_IMG, 1024, 0, stream>>>(iou, props, selScore, out);
}